// NousAlgebraBranch_85126251807081
// MI455X (gfx1250) — compile-verified
//
#include <hip/hip_runtime.h>

// NousAlgebraBranch: vectorized quadratic solver, out[i] = [[re0,im0],[re1,im1]].
// Memory-bound streaming kernel: 28 B/elem, ~20us floor at 23.3 TB/s.
// CDNA5 path: async global->LDS staging (ASYNCcnt double buffering) + NT b128 stores.

typedef __attribute__((ext_vector_type(4))) float f32x4;
typedef __attribute__((ext_vector_type(4))) int   i32x4;

#define BLOCK 256
#define VEC   4
#define TILE  (BLOCK * VEC)   // 1024 elements per block-tile

#if defined(__HIP_DEVICE_COMPILE__) && \
    __has_builtin(__builtin_amdgcn_global_load_async_to_lds_b128)
#define HAS_ASYNC 1
#else
#define HAS_ASYNC 0
#endif

__device__ __forceinline__ f32x4 solve1(float a, float b, float c, float wd, float wr) {
    float D   = fmaf(b, b, -4.0f * (a * c)) * wd;     // (b*b - 4ac) * W_disc
    float inv = 1.0f / (2.0f * a);                    // shared reciprocal of 2a
    float r   = (-b * inv) * wr;                      // -b/(2a) * W_root
    float off = sqrtf(fabsf(D)) * inv;                // sqrt(|D|)/(2a)
    float s0  = r - off, s1 = r + off;
    float lo  = fminf(s0, s1), hi = fmaxf(s0, s1);
    bool  m   = (D >= 0.0f);
    f32x4 o;
    o.x = m ? lo   : r;      // root0 real
    o.y = m ? 0.0f : -off;   // root0 imag
    o.z = m ? hi   : r;      // root1 real
    o.w = m ? 0.0f : off;    // root1 imag
    return o;
}

#if HAS_ASYNC
__device__ __forceinline__ void async_cp16(const float* g, void* l) {
    // 16 bytes per lane, HBM -> LDS via the async engine (ASYNCcnt++).
    // Builtin prototype (from hipcc diagnostic): v4i32 pointers.
    __builtin_amdgcn_global_load_async_to_lds_b128(
        (__attribute__((address_space(1))) i32x4*)(float*)g,
        (__attribute__((address_space(3))) i32x4*)l,
        0, 0);
}
#endif

__global__ __launch_bounds__(BLOCK) void quad_roots_kernel(
    const float* __restrict__ a, const float* __restrict__ b,
    const float* __restrict__ c,
    const float* __restrict__ Wd, const float* __restrict__ Wr,
    float* __restrict__ out, long long n)
{
    const float wd = Wd[0];     // uniform -> s_load
    const float wr = Wr[0];
    f32x4* __restrict__ out4 = (f32x4*)out;
    const int t = threadIdx.x;
    const long long nTiles  = n / TILE;
    const long long gstride = gridDim.x;

#if HAS_ASYNC
    // Double-buffered async staging. Each thread owns its private 3x16B slot
    // per buffer, so no workgroup barriers are needed: s_wait_asynccnt alone
    // orders each wave's own copies before its own ds_loads.
    __shared__ f32x4 lds[2][3][BLOCK];   // 24 KB / block

    long long tile = blockIdx.x;
    int p = 0;
    if (tile < nTiles) {
        long long e0 = tile * TILE + 4LL * t;
        async_cp16(a + e0, &lds[0][0][t]);
        async_cp16(b + e0, &lds[0][1][t]);
        async_cp16(c + e0, &lds[0][2][t]);
    }
    for (; tile < nTiles; tile += gstride) {
        long long nxt = tile + gstride;
        if (nxt < nTiles) {
            long long e1 = nxt * TILE + 4LL * t;
            int q = p ^ 1;
            async_cp16(a + e1, &lds[q][0][t]);
            async_cp16(b + e1, &lds[q][1][t]);
            async_cp16(c + e1, &lds[q][2][t]);
            // 6 outstanding; wait until only next tile's 3 remain in flight.
            asm volatile("s_wait_asynccnt 3" ::: "memory");
        } else {
            asm volatile("s_wait_asynccnt 0" ::: "memory");
        }
        f32x4 av = lds[p][0][t];   // ds_load_b128
        f32x4 bv = lds[p][1][t];
        f32x4 cv = lds[p][2][t];
        long long e0 = tile * TILE + 4LL * t;
#pragma unroll
        for (int k = 0; k < VEC; ++k) {
            f32x4 o = solve1(av[k], bv[k], cv[k], wd, wr);
            __builtin_nontemporal_store(o, &out4[e0 + k]);   // NT b128 store
        }
        p ^= 1;
    }
#else
    // Fallback: fully coalesced NT loads (b32, 128B/wave-instr) with
    // global_prefetch_b8 read-ahead; strided-by-BLOCK element ownership makes
    // the b128 output stores contiguous per wave (512B/instr).
    for (long long tile = blockIdx.x; tile < nTiles; tile += gstride) {
        long long base = tile * TILE;
        long long pf   = (tile + gstride) * TILE;
        if (pf < n) {
            __builtin_prefetch(&a[pf + t], 0, 0);
            __builtin_prefetch(&b[pf + t], 0, 0);
            __builtin_prefetch(&c[pf + t], 0, 0);
        }
#pragma unroll
        for (int k = 0; k < VEC; ++k) {
            long long e = base + t + BLOCK * k;
            float av = __builtin_nontemporal_load(&a[e]);
            float bv = __builtin_nontemporal_load(&b[e]);
            float cv = __builtin_nontemporal_load(&c[e]);
            f32x4 o = solve1(av, bv, cv, wd, wr);
            __builtin_nontemporal_store(o, &out4[e]);
        }
    }
#endif

    // Tail for n not divisible by TILE (empty for N = 2^24).
    for (long long e = nTiles * TILE + (long long)blockIdx.x * BLOCK + t;
         e < n; e += gstride * BLOCK) {
        f32x4 o = solve1(a[e], b[e], c[e], wd, wr);
        __builtin_nontemporal_store(o, &out4[e]);
    }
}

extern "C" void kernel_launch(void* const* d_in, const int* in_sizes, int n_in,
                              void* d_out, int out_size, void* d_ws, size_t ws_size,
                              hipStream_t stream) {
    const float* a  = (const float*)d_in[0];
    const float* b  = (const float*)d_in[1];
    const float* c  = (const float*)d_in[2];
    const float* Wd = (const float*)d_in[3];
    const float* Wr = (const float*)d_in[4];
    float* out = (float*)d_out;

    long long n     = (long long)in_sizes[0];
    long long tiles = n / TILE;
    long long blk   = tiles < 1 ? 1 : (tiles > 4096 ? 4096 : tiles);
    // 4096 blocks x 8 waves, each block walks ~4 tiles -> double buffering
    // active, tens of MB in flight, enough to saturate 23.3 TB/s.
    quad_roots_kernel<<<(int)blk, BLOCK, 0, stream>>>(a, b, c, Wd, Wr, out, n);
}